// GAT_6605659701282
// MI455X (gfx1250) — compile-verified
//
#include <hip/hip_runtime.h>
#include <math.h>

#define N_NODES 20000
#define HEADS 8
#define HID 32
#define F1 256          /* IN_CH and HEADS*HID both = 256 */
#define OUT_CH 40
#define NEG_SLOPE 0.2f

typedef __attribute__((ext_vector_type(2))) float v2f;
typedef __attribute__((ext_vector_type(8))) float v8f;

// monotone float <-> uint mapping so unsigned atomicMax == float max
__device__ __forceinline__ unsigned fenc(float f) {
    unsigned u = __float_as_uint(f);
    return (u & 0x80000000u) ? ~u : (u | 0x80000000u);
}
__device__ __forceinline__ float fdec(unsigned u) {
    return (u & 0x80000000u) ? __uint_as_float(u & 0x7FFFFFFFu)
                             : __uint_as_float(~u);
}

// ---------------- GEMM1: H[20000,256] = x[20000,256] @ W1[256,256] ----------
// 256 threads = 8 waves; block stages a 16x256 row-tile of x into LDS (float4,
// fully coalesced). Each wave owns 32 consecutive output columns (two 16x16
// WMMA tiles sharing one A fragment per k-step -> B loads cover a full 128B
// line, DS traffic halved, two independent WMMA chains in flight).
__global__ void __launch_bounds__(256)
gat_gemm1_wmma(const float* __restrict__ x, const float* __restrict__ W,
               float* __restrict__ H) {
    __shared__ __align__(16) float xs[16 * F1];   // 16 KB
    const int rowBase = blockIdx.x * 16;
    const int tid = threadIdx.x;
    const float4* xrow4 = (const float4*)(x + (size_t)rowBase * F1);
    float4* xs4 = (float4*)xs;
#pragma unroll
    for (int j = 0; j < 4; ++j) xs4[j * 256 + tid] = xrow4[j * 256 + tid];
    __syncthreads();

    const int wave  = tid >> 5;
    const int lane  = tid & 31;
    const int l16   = lane & 15;
    const int khalf = lane >> 4;                  // 0: K={0,1}  1: K={2,3}
    const float* arow = xs + l16 * F1 + 2 * khalf;
    const int col0 = wave * 32 + l16;
    const int col1 = col0 + 16;

    v8f c0 = {}, c1 = {};
#pragma unroll 8
    for (int k0 = 0; k0 < F1; k0 += 4) {
        v2f a, b0, b1;
        a.x = arow[k0];
        a.y = arow[k0 + 1];
        const float* wr0 = W + (k0 + 2 * khalf) * F1;
        const float* wr1 = wr0 + F1;
        b0.x = wr0[col0]; b0.y = wr1[col0];
        b1.x = wr0[col1]; b1.y = wr1[col1];
        c0 = __builtin_amdgcn_wmma_f32_16x16x4_f32(false, a, false, b0,
                                                   (short)0, c0, false, false);
        c1 = __builtin_amdgcn_wmma_f32_16x16x4_f32(false, a, false, b1,
                                                   (short)0, c1, false, false);
    }
#pragma unroll
    for (int r = 0; r < 8; ++r) {
        float* hrow = H + (size_t)(rowBase + r + 8 * khalf) * F1;
        hrow[col0] = c0[r];
        hrow[col1] = c1[r];
    }
}

// ---------------- GEMM2: g[20000,40] = h2[20000,256] @ W2[256,40] -----------
__global__ void __launch_bounds__(256)
gat_gemm2_wmma(const float* __restrict__ h2, const float* __restrict__ W,
               float* __restrict__ g) {
    __shared__ __align__(16) float xs[16 * F1];
    const int rowBase = blockIdx.x * 16;
    const int tid = threadIdx.x;
    const float4* xrow4 = (const float4*)(h2 + (size_t)rowBase * F1);
    float4* xs4 = (float4*)xs;
#pragma unroll
    for (int j = 0; j < 4; ++j) xs4[j * 256 + tid] = xrow4[j * 256 + tid];
    __syncthreads();

    const int wave = tid >> 5;
    if (wave < 3) {                               // 3 col tiles cover 40 cols
        const int lane  = tid & 31;
        const int l16   = lane & 15;
        const int khalf = lane >> 4;
        const int col   = wave * 16 + l16;
        const bool valid = col < OUT_CH;
        const float* arow = xs + l16 * F1 + 2 * khalf;
        v8f c = {};
#pragma unroll 8
        for (int k0 = 0; k0 < F1; k0 += 4) {
            v2f a, b;
            a.x = arow[k0];
            a.y = arow[k0 + 1];
            b.x = valid ? W[(k0 + 2 * khalf) * OUT_CH + col] : 0.0f;
            b.y = valid ? W[(k0 + 2 * khalf + 1) * OUT_CH + col] : 0.0f;
            c = __builtin_amdgcn_wmma_f32_16x16x4_f32(false, a, false, b,
                                                      (short)0, c, false, false);
        }
        if (valid) {
#pragma unroll
            for (int r = 0; r < 8; ++r)
                g[(size_t)(rowBase + r + 8 * khalf) * OUT_CH + col] = c[r];
        }
    }
}

// ---------------- per-node attention logits ---------------------------------
__global__ void gat_alpha1(const float* __restrict__ H,
                           const float* __restrict__ a1s,
                           const float* __restrict__ a1d,
                           float* __restrict__ aS, float* __restrict__ aD) {
    int t = blockIdx.x * blockDim.x + threadIdx.x;
    if (t >= N_NODES * HEADS) return;
    int n = t >> 3, hd = t & 7;
    const float* hp = H + (size_t)n * F1 + hd * HID;
    const float* sv = a1s + hd * HID;
    const float* dv = a1d + hd * HID;
    float vs = 0.f, vd = 0.f;
#pragma unroll
    for (int c = 0; c < HID; ++c) { float h = hp[c]; vs += h * sv[c]; vd += h * dv[c]; }
    aS[t] = vs; aD[t] = vd;
}

__global__ void gat_alpha2(const float* __restrict__ g,
                           const float* __restrict__ a2s,
                           const float* __restrict__ a2d,
                           float* __restrict__ aS, float* __restrict__ aD) {
    int n = blockIdx.x * blockDim.x + threadIdx.x;
    if (n >= N_NODES) return;
    const float* gp = g + (size_t)n * OUT_CH;
    float vs = 0.f, vd = 0.f;
#pragma unroll
    for (int c = 0; c < OUT_CH; ++c) { float h = gp[c]; vs += h * a2s[c]; vd += h * a2d[c]; }
    aS[n] = vs; aD[n] = vd;
}

// ---------------- edge pass A: leaky-relu logit + segment max ---------------
__global__ void gat_edge_pre1(const int* __restrict__ ei, int E0, int nEH,
                              const float* __restrict__ aS, const float* __restrict__ aD,
                              float* __restrict__ ebuf, unsigned* __restrict__ m) {
    int t = blockIdx.x * blockDim.x + threadIdx.x;
    if (t >= nEH) return;
    int edge = t >> 3, hd = t & 7;
    int s = (edge < E0) ? ei[edge]      : (edge - E0);
    int d = (edge < E0) ? ei[E0 + edge] : (edge - E0);
    float e = aS[s * HEADS + hd] + aD[d * HEADS + hd];
    e = (e > 0.f) ? e : NEG_SLOPE * e;
    ebuf[t] = e;
    atomicMax(&m[d * HEADS + hd], fenc(e));
}

__global__ void gat_edge_pre2(const int* __restrict__ ei, int E0, int Etot,
                              const float* __restrict__ aS, const float* __restrict__ aD,
                              float* __restrict__ ebuf, unsigned* __restrict__ m) {
    int edge = blockIdx.x * blockDim.x + threadIdx.x;
    if (edge >= Etot) return;
    int s = (edge < E0) ? ei[edge]      : (edge - E0);
    int d = (edge < E0) ? ei[E0 + edge] : (edge - E0);
    float e = aS[s] + aD[d];
    e = (e > 0.f) ? e : NEG_SLOPE * e;
    ebuf[edge] = e;
    atomicMax(&m[d], fenc(e));
}

// ---------------- edge pass B: exp(e - m[dst]), segment sum -----------------
__global__ void gat_edge_exp1(const int* __restrict__ ei, int E0, int nEH,
                              const unsigned* __restrict__ m,
                              float* __restrict__ ebuf, float* __restrict__ denom) {
    int t = blockIdx.x * blockDim.x + threadIdx.x;
    if (t >= nEH) return;
    int edge = t >> 3, hd = t & 7;
    int d = (edge < E0) ? ei[E0 + edge] : (edge - E0);
    float ex = __expf(ebuf[t] - fdec(m[d * HEADS + hd]));
    ebuf[t] = ex;
    atomicAdd(&denom[d * HEADS + hd], ex);
}

__global__ void gat_edge_exp2(const int* __restrict__ ei, int E0, int Etot,
                              const unsigned* __restrict__ m,
                              float* __restrict__ ebuf, float* __restrict__ denom) {
    int edge = blockIdx.x * blockDim.x + threadIdx.x;
    if (edge >= Etot) return;
    int d = (edge < E0) ? ei[E0 + edge] : (edge - E0);
    float ex = __expf(ebuf[edge] - fdec(m[d]));
    ebuf[edge] = ex;
    atomicAdd(&denom[d], ex);
}

// ---------------- edge pass C: weighted scatter aggregation -----------------
// one wave handles one (edge, head); lane = channel -> 32 consecutive f32
// atomics per wave (coalesced L2 burst); edge metadata loads are wave-uniform.
__global__ void gat_edge_agg1(const int* __restrict__ ei, int E0, int total,
                              const float* __restrict__ ebuf, const float* __restrict__ denom,
                              const float* __restrict__ H, float* __restrict__ out) {
    int t = blockIdx.x * blockDim.x + threadIdx.x;
    if (t >= total) return;                       // total = Etot*HEADS*HID
    int c = t & 31;
    int eh = t >> 5;
    int edge = eh >> 3, hd = eh & 7;
    int s = (edge < E0) ? ei[edge]      : (edge - E0);
    int d = (edge < E0) ? ei[E0 + edge] : (edge - E0);
    float alpha = ebuf[eh] / (denom[d * HEADS + hd] + 1e-16f);
    atomicAdd(&out[(size_t)d * F1 + hd * HID + c],
              H[(size_t)s * F1 + hd * HID + c] * alpha);
}

__global__ void gat_edge_agg2(const int* __restrict__ ei, int E0, int total,
                              const float* __restrict__ ebuf, const float* __restrict__ denom,
                              const float* __restrict__ g, float* __restrict__ out) {
    int t = blockIdx.x * blockDim.x + threadIdx.x;
    if (t >= total) return;                       // total = Etot*OUT_CH
    int edge = t / OUT_CH;
    int c = t - edge * OUT_CH;
    int s = (edge < E0) ? ei[edge]      : (edge - E0);
    int d = (edge < E0) ? ei[E0 + edge] : (edge - E0);
    float alpha = ebuf[edge] / (denom[d] + 1e-16f);
    atomicAdd(&out[(size_t)d * OUT_CH + c], g[(size_t)s * OUT_CH + c] * alpha);
}

// ---------------- bias + ELU between layers ---------------------------------
__global__ void gat_elu_bias(const float* __restrict__ acc, const float* __restrict__ b1,
                             float* __restrict__ h2) {
    int t = blockIdx.x * blockDim.x + threadIdx.x;
    if (t >= N_NODES * F1) return;
    float v = acc[t] + b1[t & (F1 - 1)];
    h2[t] = (v > 0.f) ? v : (__expf(v) - 1.f);
}

// ---------------- bias + log_softmax ----------------------------------------
__global__ void gat_final(const float* __restrict__ out2, const float* __restrict__ b2,
                          float* __restrict__ y) {
    int n = blockIdx.x * blockDim.x + threadIdx.x;
    if (n >= N_NODES) return;
    float v[OUT_CH];
    float mx = -3.0e38f;
#pragma unroll
    for (int c = 0; c < OUT_CH; ++c) {
        v[c] = out2[(size_t)n * OUT_CH + c] + b2[c];
        mx = fmaxf(mx, v[c]);
    }
    float s = 0.f;
#pragma unroll
    for (int c = 0; c < OUT_CH; ++c) s += __expf(v[c] - mx);
    float lse = mx + __logf(s);
#pragma unroll
    for (int c = 0; c < OUT_CH; ++c) y[(size_t)n * OUT_CH + c] = v[c] - lse;
}

extern "C" void kernel_launch(void* const* d_in, const int* in_sizes, int n_in,
                              void* d_out, int out_size, void* d_ws, size_t ws_size,
                              hipStream_t stream) {
    const float* x   = (const float*)d_in[0];
    const int*   ei  = (const int*)d_in[1];
    const float* W1  = (const float*)d_in[2];
    const float* a1s = (const float*)d_in[3];
    const float* a1d = (const float*)d_in[4];
    const float* b1  = (const float*)d_in[5];
    const float* W2  = (const float*)d_in[6];
    const float* a2s = (const float*)d_in[7];
    const float* a2d = (const float*)d_in[8];
    const float* b2  = (const float*)d_in[9];
    float* y = (float*)d_out;

    const int E0   = in_sizes[1] / 2;
    const int Etot = E0 + N_NODES;

    // ---- workspace layout (floats). Zero-init region first, one memset. ----
    float* ws = (float*)d_ws;
    size_t off = 0;
    unsigned* m1   = (unsigned*)(ws + off); off += (size_t)N_NODES * HEADS;
    float* denom1  = ws + off;              off += (size_t)N_NODES * HEADS;
    float* out1    = ws + off;              off += (size_t)N_NODES * F1;
    unsigned* m2   = (unsigned*)(ws + off); off += N_NODES;
    float* denom2  = ws + off;              off += N_NODES;
    float* out2    = ws + off;              off += (size_t)N_NODES * OUT_CH;
    const size_t zeroFloats = off;
    float* H1  = ws + off; off += (size_t)N_NODES * F1;
    float* aS1 = ws + off; off += (size_t)N_NODES * HEADS;
    float* aD1 = ws + off; off += (size_t)N_NODES * HEADS;
    float* e1  = ws + off; off += (size_t)Etot * HEADS;
    float* h2  = ws + off; off += (size_t)N_NODES * F1;
    float* g   = ws + off; off += (size_t)N_NODES * OUT_CH;
    float* aS2 = ws + off; off += N_NODES;
    float* aD2 = ws + off; off += N_NODES;
    float* e2  = ws + off; off += Etot;
    (void)ws_size; (void)n_in; (void)out_size;

    hipMemsetAsync(d_ws, 0, zeroFloats * sizeof(float), stream);

    // ---- layer 1 ----
    gat_gemm1_wmma<<<N_NODES / 16, 256, 0, stream>>>(x, W1, H1);
    gat_alpha1<<<(N_NODES * HEADS + 255) / 256, 256, 0, stream>>>(H1, a1s, a1d, aS1, aD1);
    const int nEH = Etot * HEADS;
    gat_edge_pre1<<<(nEH + 255) / 256, 256, 0, stream>>>(ei, E0, nEH, aS1, aD1, e1, m1);
    gat_edge_exp1<<<(nEH + 255) / 256, 256, 0, stream>>>(ei, E0, nEH, m1, e1, denom1);
    const int nAgg1 = nEH * HID;                       // 87.04M lanes
    gat_edge_agg1<<<(nAgg1 + 255) / 256, 256, 0, stream>>>(ei, E0, nAgg1, e1, denom1, H1, out1);
    gat_elu_bias<<<(N_NODES * F1 + 255) / 256, 256, 0, stream>>>(out1, b1, h2);

    // ---- layer 2 ----
    gat_gemm2_wmma<<<N_NODES / 16, 256, 0, stream>>>(h2, W2, g);
    gat_alpha2<<<(N_NODES + 255) / 256, 256, 0, stream>>>(g, a2s, a2d, aS2, aD2);
    gat_edge_pre2<<<(Etot + 255) / 256, 256, 0, stream>>>(ei, E0, Etot, aS2, aD2, e2, m2);
    gat_edge_exp2<<<(Etot + 255) / 256, 256, 0, stream>>>(ei, E0, Etot, m2, e2, denom2);
    const int nAgg2 = Etot * OUT_CH;
    gat_edge_agg2<<<(nAgg2 + 255) / 256, 256, 0, stream>>>(ei, E0, nAgg2, e2, denom2, g, out2);
    gat_final<<<(N_NODES + 255) / 256, 256, 0, stream>>>(out2, b2, y);
}